// MyAttn_30820685316844
// MI455X (gfx1250) — compile-verified
//
#include <hip/hip_runtime.h>
#include <math.h>

#define NN 1024      // nodes
#define HH 128       // hidden
#define NHEAD 8
#define DHD 16       // head dim

typedef float v2f __attribute__((ext_vector_type(2)));
typedef float v8f __attribute__((ext_vector_type(8)));

// ---------------- utility: zero int buffer ----------------
__global__ void zero_ints(int* p, int n) {
    for (int i = blockIdx.x * blockDim.x + threadIdx.x; i < n; i += gridDim.x * blockDim.x)
        p[i] = 0;
}

// ---------------- adjacency + dst histogram ----------------
__global__ void build_adj_hist(const int* __restrict__ src, const int* __restrict__ dst,
                               int E, int* __restrict__ adj, int* __restrict__ counts) {
    int e = blockIdx.x * blockDim.x + threadIdx.x;
    if (e < E) {
        adj[(size_t)src[e] * NN + dst[e]] = 1;   // benign race: all write 1
        atomicAdd(&counts[dst[e]], 1);
    }
}

// ---------------- single-block inclusive scan over 1024 counts ----------------
__global__ __launch_bounds__(NN) void scan_offsets(const int* __restrict__ counts,
                                                   int* __restrict__ rowStart,
                                                   int* __restrict__ cursor) {
    __shared__ int buf[NN];
    int t = threadIdx.x;
    buf[t] = counts[t];
    __syncthreads();
    for (int off = 1; off < NN; off <<= 1) {
        int x = (t >= off) ? buf[t - off] : 0;
        __syncthreads();
        buf[t] += x;
        __syncthreads();
    }
    rowStart[t + 1] = buf[t];
    int excl = (t == 0) ? 0 : buf[t - 1];
    cursor[t] = excl;
    if (t == 0) rowStart[0] = 0;
}

// ---------------- scatter edge ids into dst buckets ----------------
__global__ void scatter_edges(const int* __restrict__ dst, int E,
                              int* __restrict__ cursor, int* __restrict__ sortedE) {
    int e = blockIdx.x * blockDim.x + threadIdx.x;
    if (e < E) {
        int p = atomicAdd(&cursor[dst[e]], 1);
        sortedE[p] = e;
    }
}

// ---------------- fp32 WMMA GEMM: C[M,128] = A[M,128] @ W[128,128]^T (+bias) ----
// grid.x = M/16 ; block = 256 (8 waves, one 16-col tile each => N=128)
__global__ __launch_bounds__(256) void gemm_nt_128(const float* __restrict__ A,
                                                   const float* __restrict__ W,
                                                   const float* __restrict__ bias,
                                                   float* __restrict__ C, int M) {
    const int mtile = blockIdx.x;
    const int wave  = threadIdx.x >> 5;   // 0..7 -> column tile
    const int lane  = threadIdx.x & 31;
    const int half  = lane >> 4;          // K sub-group (0: K0/K1, 1: K2/K3)
    const int l     = lane & 15;

    const int row0 = mtile * 16;
    const int col0 = wave * 16;

    const float* Arow = A + (size_t)(row0 + l) * HH;   // A[row0+l][*]
    const float* Wrow = W + (size_t)(col0 + l) * HH;   // B[k][n] = W[n][k]

    v8f acc = {};
    #pragma unroll 4
    for (int k0 = 0; k0 < HH; k0 += 4) {
        v2f a = *(const v2f*)(Arow + k0 + 2 * half);
        v2f b = *(const v2f*)(Wrow + k0 + 2 * half);
        acc = __builtin_amdgcn_wmma_f32_16x16x4_f32(
                  false, a, false, b, (short)0, acc, false, false);
    }

    const float bb = bias ? bias[col0 + l] : 0.0f;
    float* Cp = C + (size_t)(row0 + half * 8) * HH + col0 + l;
    #pragma unroll
    for (int i = 0; i < 8; ++i)
        Cp[(size_t)i * HH] = acc[i] + bb;
}

// ---------------- fused K+V projection with async-to-LDS A staging ----------------
// KE[M,128] = A @ Wk^T ; VE[M,128] = A @ Wv^T   (A tile read once per block)
// grid.x = M/16 ; block = 256 (8 waves; wave w owns col tile w for BOTH outputs)
#define AT_STRIDE 132   // pad: bank = (4*l + c) mod 64 -> conflict-free b64 reads
__global__ __launch_bounds__(256) void proj_kv(const float* __restrict__ A,
                                               const float* __restrict__ Wk,
                                               const float* __restrict__ Wv,
                                               float* __restrict__ KE,
                                               float* __restrict__ VE, int M) {
    __shared__ float At[16][AT_STRIDE];

    const int t    = threadIdx.x;
    const int wave = t >> 5;
    const int lane = t & 31;
    const int half = lane >> 4;
    const int l    = lane & 15;

    const int row0 = blockIdx.x * 16;
    const int col0 = wave * 16;

    // ---- async stage of the 16x128 A tile (512 x 16B chunks, 2 per thread) ----
    {
        const float* gbase = A + (size_t)row0 * HH;
        #pragma unroll
        for (int i = 0; i < 2; ++i) {
            const int c   = t + i * 256;   // chunk id 0..511
            const int r   = c >> 5;        // row 0..15
            const int c16 = c & 31;        // 16-byte column chunk
            // low 32 bits of a generic LDS pointer == LDS byte offset (aperture rule)
            const unsigned int lds_off = (unsigned int)(size_t)(&At[r][c16 * 4]);
            const void* g = (const void*)(gbase + (size_t)r * HH + c16 * 4);
            asm volatile("global_load_async_to_lds_b128 %0, %1, off"
                         :: "v"(lds_off), "v"(g) : "memory");
        }
        asm volatile("s_wait_asynccnt 0x0" ::: "memory");
    }
    __syncthreads();

    const float* WkRow = Wk + (size_t)(col0 + l) * HH;
    const float* WvRow = Wv + (size_t)(col0 + l) * HH;

    v8f acck = {};
    v8f accv = {};
    #pragma unroll 4
    for (int k0 = 0; k0 < HH; k0 += 4) {
        v2f a  = *(const v2f*)(&At[l][k0 + 2 * half]);      // ds_load_b64, bank-clean
        v2f bk = *(const v2f*)(WkRow + k0 + 2 * half);
        v2f bv = *(const v2f*)(WvRow + k0 + 2 * half);
        acck = __builtin_amdgcn_wmma_f32_16x16x4_f32(
                   false, a, false, bk, (short)0, acck, false, false);
        accv = __builtin_amdgcn_wmma_f32_16x16x4_f32(
                   false, a, false, bv, (short)0, accv, false, false);
    }

    float* Kp = KE + (size_t)(row0 + half * 8) * HH + col0 + l;
    float* Vp = VE + (size_t)(row0 + half * 8) * HH + col0 + l;
    #pragma unroll
    for (int i = 0; i < 8; ++i) {
        Kp[(size_t)i * HH] = acck[i];
        Vp[(size_t)i * HH] = accv[i];
    }
}

// ---------------- sparse masked attention: one block per dst node ----------------
__global__ __launch_bounds__(256) void attn_kernel(const float* __restrict__ Q,
                                                   const float* __restrict__ KE,
                                                   const float* __restrict__ VE,
                                                   const int* __restrict__ adj,
                                                   const int* __restrict__ rowStart,
                                                   const int* __restrict__ sortedE,
                                                   const int* __restrict__ esrc,
                                                   const float* __restrict__ bv,
                                                   float* __restrict__ out) {
    __shared__ float sc[NHEAD][NN];   // per-head score/weight table (32 KB)
    __shared__ float qv[HH];

    const int b = blockIdx.x;
    const int t = threadIdx.x;

    if (t < HH) qv[t] = Q[(size_t)b * HH + t];
    for (int i = t; i < NHEAD * NN; i += 256) (&sc[0][0])[i] = 0.0f;
    __syncthreads();

    const int e0 = rowStart[b], e1 = rowStart[b + 1];
    const int m  = e1 - e0;

    // score accumulation: each (edge, head) pair -> dot over 16 dims
    for (int idx = t; idx < m * NHEAD; idx += 256) {
        const int ei = idx >> 3;
        const int nh = idx & 7;
        const int e  = sortedE[e0 + ei];
        const int s  = esrc[e];
        const float* kp = KE + (size_t)e * HH + nh * DHD;
        const float* qp = qv + nh * DHD;
        float d = 0.0f;
        #pragma unroll
        for (int j = 0; j < DHD; ++j) d = fmaf(qp[j], kp[j], d);
        atomicAdd(&sc[nh][s], d);     // ds_add_f32
    }
    __syncthreads();

    // masked softmax: wave w handles head w (wave32-native reductions)
    {
        const int nh   = t >> 5;
        const int lane = t & 31;
        const int* arow = adj + (size_t)b * NN;

        float mx = -INFINITY;
        for (int s = lane; s < NN; s += 32) {
            const bool ok = (s == b) || (arow[s] != 0);
            const float v = ok ? sc[nh][s] * 0.25f : -INFINITY;  // 1/sqrt(DH)
            sc[nh][s] = v;
            mx = fmaxf(mx, v);
        }
        #pragma unroll
        for (int o = 16; o; o >>= 1) mx = fmaxf(mx, __shfl_xor(mx, o, 32));

        float sum = 0.0f;
        for (int s = lane; s < NN; s += 32) {
            const float v = sc[nh][s];
            const float e = (v == -INFINITY) ? 0.0f : __expf(v - mx);
            sc[nh][s] = e;
            sum += e;
        }
        #pragma unroll
        for (int o = 16; o; o >>= 1) sum += __shfl_xor(sum, o, 32);

        const float inv = 1.0f / sum;
        for (int s = lane; s < NN; s += 32) sc[nh][s] *= inv;
    }
    __syncthreads();

    // weighted value gather: out[b] = bv + sum_e attn[src_e] * VE[e]
    if (t < HH) {
        const int nh = t >> 4;
        float acc = bv[t];
        for (int ei = 0; ei < m; ++ei) {
            const int e = sortedE[e0 + ei];
            const int s = esrc[e];
            acc = fmaf(sc[nh][s], VE[(size_t)e * HH + t], acc);
        }
        out[(size_t)b * HH + t] = acc;
    }
}

// =============================== launcher ===============================
extern "C" void kernel_launch(void* const* d_in, const int* in_sizes, int n_in,
                              void* d_out, int out_size, void* d_ws, size_t ws_size,
                              hipStream_t stream) {
    const float* x          = (const float*)d_in[0];
    const int*   edge_index = (const int*)d_in[1];
    const float* edge_attr  = (const float*)d_in[2];
    /* d_in[3] = batch (unused: single graph) */
    const float* in_proj_w  = (const float*)d_in[4];
    const float* in_proj_b  = (const float*)d_in[5];
    const float* out_w      = (const float*)d_in[6];
    const float* out_b      = (const float*)d_in[7];
    float* out              = (float*)d_out;

    const int E = in_sizes[1] / 2;                  // 32768
    const int* esrc = edge_index;
    const int* edst = edge_index + E;

    // ---- workspace carve-up (≈37 MB) ----
    char* base = (char*)d_ws;
    size_t off = 0;
    auto take = [&](size_t bytes) { void* p = base + off; off = (off + bytes + 255) & ~(size_t)255; return p; };
    int*   adj      = (int*)take((size_t)NN * NN * sizeof(int));
    int*   counts   = (int*)take((NN + 1) * sizeof(int));
    int*   rowStart = (int*)take((NN + 1) * sizeof(int));
    int*   cursor   = (int*)take(NN * sizeof(int));
    int*   sortedE  = (int*)take((size_t)E * sizeof(int));
    float* Qw       = (float*)take((size_t)NN * HH * sizeof(float));
    float* KE       = (float*)take((size_t)E * HH * sizeof(float));
    float* VE       = (float*)take((size_t)E * HH * sizeof(float));
    float* attnOut  = (float*)take((size_t)NN * HH * sizeof(float));

    const float* wq = in_proj_w;
    const float* wk = in_proj_w + (size_t)HH * HH;
    const float* wv = in_proj_w + (size_t)2 * HH * HH;
    const float* bq = in_proj_b;
    const float* bv = in_proj_b + 2 * HH;

    // 1) clear adjacency + histogram
    zero_ints<<<512, 256, 0, stream>>>(adj, NN * NN);
    zero_ints<<<4, 256, 0, stream>>>(counts, NN + 1);

    // 2) adjacency + dst histogram
    build_adj_hist<<<(E + 255) / 256, 256, 0, stream>>>(esrc, edst, E, adj, counts);

    // 3) CSR offsets
    scan_offsets<<<1, NN, 0, stream>>>(counts, rowStart, cursor);

    // 4) bucket edges by dst
    scatter_edges<<<(E + 255) / 256, 256, 0, stream>>>(edst, E, cursor, sortedE);

    // 5) WMMA projections (fp32 16x16x4)
    gemm_nt_128<<<NN / 16, 256, 0, stream>>>(x, wq, bq, Qw, NN);            // Q = x Wq^T + bq
    proj_kv<<<E / 16, 256, 0, stream>>>(edge_attr, wk, wv, KE, VE, E);      // fused K+V proj

    // 6) sparse masked attention (bk drops out of softmax; bv enters via sum(attn)=1)
    attn_kernel<<<NN, 256, 0, stream>>>(Qw, KE, VE, adj, rowStart, sortedE, esrc, bv, attnOut);

    // 7) output projection
    gemm_nt_128<<<NN / 16, 256, 0, stream>>>(attnOut, out_w, out_b, out, NN);
}